// SubmanifoldSparseConv_13365938225624
// MI455X (gfx1250) — compile-verified
//
#include <hip/hip_runtime.h>
#include <hip/hip_bf16.h>

typedef __attribute__((ext_vector_type(2))) float v2f;
typedef __attribute__((ext_vector_type(8))) float v8f;

#define CIN 32
#define COUT 32
#define K3 27
#define MKEY 1024
#define TS_LOG2 19
#define TS (1u << TS_LOG2)
#define EMPTY_KEY 0xFFFFFFFFu

__device__ __forceinline__ uint32_t hash_key(uint32_t k) {
    return (k * 2654435761u) >> (32 - TS_LOG2);
}

__global__ void init_table_kernel(uint32_t* __restrict__ keys) {
    unsigned i = blockIdx.x * blockDim.x + threadIdx.x;
    if (i < TS) keys[i] = EMPTY_KEY;
}

__global__ void build_table_kernel(const float* __restrict__ inp_pos,
                                   uint32_t* __restrict__ keys,
                                   int* __restrict__ vals, int n) {
    int i = blockIdx.x * blockDim.x + threadIdx.x;
    if (i >= n) return;
    int x = (int)floorf(inp_pos[i * 3 + 0]);
    int y = (int)floorf(inp_pos[i * 3 + 1]);
    int z = (int)floorf(inp_pos[i * 3 + 2]);
    uint32_t key = (uint32_t)((x * MKEY + y) * MKEY + z);
    uint32_t h = hash_key(key);
    for (;;) {
        uint32_t prev = atomicCAS((unsigned int*)&keys[h], EMPTY_KEY, key);
        if (prev == EMPTY_KEY || prev == key) { vals[h] = i; break; }
        h = (h + 1) & (TS - 1);
    }
}

__device__ __forceinline__ int probe_table(const uint32_t* __restrict__ hkeys,
                                           const int* __restrict__ hvals,
                                           uint32_t key, bool valid) {
    int src = -1;
    if (valid && key != EMPTY_KEY) {
        uint32_t h = hash_key(key);
        for (;;) {
            uint32_t kv = hkeys[h];
            if (kv == key) { src = hvals[h]; break; }
            if (kv == EMPTY_KEY) break;
            h = (h + 1) & (TS - 1);
        }
    }
    return src;
}

// 256 threads = 8 waves; each wave computes a 32-row x 32-col output tile
// (two 16x16x4 f32 WMMA row-tiles sharing the same B fragments).
// W is staged in LDS pre-swizzled into V_WMMA_F32_16X16X4_F32 B-fragment layout:
// wlds[((k*8+s)*2+t)*32 + lane] = { W[k][4s+kh][16t+(lane&15)], W[k][4s+kh+1][...] }
// with kh = (lane<16 ? 0 : 2), mirroring the 32-bit 16x4 A layout from the ISA.
__launch_bounds__(256)
__global__ void sconv_wmma_kernel(const float* __restrict__ feat,
                                  const float* __restrict__ out_pos,
                                  const float* __restrict__ W,
                                  const uint32_t* __restrict__ hkeys,
                                  const int* __restrict__ hvals,
                                  float* __restrict__ out, int n) {
    __shared__ v2f wlds[K3 * 8 * 2 * 32];  // 13824 * 8B = 110592 B

    const int tid  = threadIdx.x;
    const int lane = tid & 31;
    const int wave = tid >> 5;

    // Pack weights into LDS B-fragments. 13824 entries / 256 threads = 54 full
    // iterations -> no divergence, EXEC stays all-ones.
    for (int e = tid; e < K3 * 8 * 2 * 32; e += 256) {
        int k  = e >> 9;          // tap
        int r  = e & 511;
        int s  = r >> 6;          // K-step (4 K-values each)
        int t  = (r >> 5) & 1;    // N-tile (cols 0-15 / 16-31)
        int ln = r & 31;
        int kk  = s * 4 + ((ln < 16) ? 0 : 2);
        int col = t * 16 + (ln & 15);
        const float* wk = W + (k * CIN + kk) * COUT + col;
        v2f v;
        v.x = wk[0];
        v.y = wk[COUT];
        wlds[e] = v;
    }
    __syncthreads();

    const int rowbase = blockIdx.x * 256 + wave * 32;
    const int row0    = rowbase + (lane & 15);        // rows 0-15 of wave tile
    const int row1    = row0 + 16;                    // rows 16-31 of wave tile
    const bool rv0 = row0 < n;
    const bool rv1 = row1 < n;
    const int p0 = rv0 ? row0 : 0;
    const int p1 = rv1 ? row1 : 0;

    int vx0 = (int)floorf(out_pos[p0 * 3 + 0]);
    int vy0 = (int)floorf(out_pos[p0 * 3 + 1]);
    int vz0 = (int)floorf(out_pos[p0 * 3 + 2]);
    int vx1 = (int)floorf(out_pos[p1 * 3 + 0]);
    int vy1 = (int)floorf(out_pos[p1 * 3 + 1]);
    int vz1 = (int)floorf(out_pos[p1 * 3 + 2]);
    const int khalf = (lane < 16) ? 0 : 2;

    v8f acc00 = {0.f, 0.f, 0.f, 0.f, 0.f, 0.f, 0.f, 0.f};  // rows 0-15, cols 0-15
    v8f acc01 = {0.f, 0.f, 0.f, 0.f, 0.f, 0.f, 0.f, 0.f};  // rows 0-15, cols 16-31
    v8f acc10 = {0.f, 0.f, 0.f, 0.f, 0.f, 0.f, 0.f, 0.f};  // rows 16-31, cols 0-15
    v8f acc11 = {0.f, 0.f, 0.f, 0.f, 0.f, 0.f, 0.f, 0.f};  // rows 16-31, cols 16-31

    for (int k = 0; k < K3; ++k) {
        const int ox = k / 9 - 1;
        const int oy = (k / 3) % 3 - 1;
        const int oz = k % 3 - 1;

        // Two independent hash probes (one per row tile) -> ILP over the
        // dependent probe-load chains.
        uint32_t key0 = (uint32_t)(((vx0 + ox) * MKEY + (vy0 + oy)) * MKEY + (vz0 + oz));
        uint32_t key1 = (uint32_t)(((vx1 + ox) * MKEY + (vy1 + oy)) * MKEY + (vz1 + oz));
        int src0 = probe_table(hkeys, hvals, key0, rv0);
        int src1 = probe_table(hkeys, hvals, key1, rv1);

        const float* fr0 = feat + (long)((src0 >= 0) ? src0 : 0) * CIN;
        const float* fr1 = feat + (long)((src1 >= 0) ? src1 : 0) * CIN;
        const bool have0 = (src0 >= 0);
        const bool have1 = (src1 >= 0);
        const v2f* wb = &wlds[k * 512];

#pragma unroll
        for (int s = 0; s < 8; ++s) {
            v2f a0, a1;
            if (have0) a0 = *(const v2f*)(fr0 + 4 * s + khalf);
            else       { a0.x = 0.f; a0.y = 0.f; }
            if (have1) a1 = *(const v2f*)(fr1 + 4 * s + khalf);
            else       { a1.x = 0.f; a1.y = 0.f; }
            v2f b0 = wb[s * 64 + lane];
            v2f b1 = wb[s * 64 + 32 + lane];
            acc00 = __builtin_amdgcn_wmma_f32_16x16x4_f32(
                false, a0, false, b0, (short)0, acc00, false, false);
            acc01 = __builtin_amdgcn_wmma_f32_16x16x4_f32(
                false, a0, false, b1, (short)0, acc01, false, false);
            acc10 = __builtin_amdgcn_wmma_f32_16x16x4_f32(
                false, a1, false, b0, (short)0, acc10, false, false);
            acc11 = __builtin_amdgcn_wmma_f32_16x16x4_f32(
                false, a1, false, b1, (short)0, acc11, false, false);
        }
    }

    // C/D layout: lane -> N = (lane&15) (+16 for *1 accs),
    // VGPR v -> M = v + (lane<16 ? 0 : 8) within each 16-row tile.
    const int crow0 = rowbase + ((lane < 16) ? 0 : 8);       // tile rows 0-15
    const int crow1 = crow0 + 16;                            // tile rows 16-31
    const int col = lane & 15;
#pragma unroll
    for (int v = 0; v < 8; ++v) {
        int r0 = crow0 + v;
        int r1 = crow1 + v;
        if (r0 < n) {
            out[(long)r0 * COUT + col]      = acc00[v];
            out[(long)r0 * COUT + 16 + col] = acc01[v];
        }
        if (r1 < n) {
            out[(long)r1 * COUT + col]      = acc10[v];
            out[(long)r1 * COUT + 16 + col] = acc11[v];
        }
    }
}

extern "C" void kernel_launch(void* const* d_in, const int* in_sizes, int n_in,
                              void* d_out, int out_size, void* d_ws, size_t ws_size,
                              hipStream_t stream) {
    const float* feat    = (const float*)d_in[0];  // [N, 32]
    const float* inp_pos = (const float*)d_in[1];  // [N, 3]
    const float* out_pos = (const float*)d_in[2];  // [N, 3]
    const float* W       = (const float*)d_in[3];  // [27, 32, 32]
    float* out = (float*)d_out;

    const int n = in_sizes[0] / CIN;

    uint32_t* hkeys = (uint32_t*)d_ws;
    int* hvals = (int*)((char*)d_ws + (size_t)TS * sizeof(uint32_t));
    // total workspace use: 2^19 * 8B = 4 MB

    init_table_kernel<<<(TS + 255) / 256, 256, 0, stream>>>(hkeys);
    build_table_kernel<<<(n + 255) / 256, 256, 0, stream>>>(inp_pos, hkeys, hvals, n);

    const int nblocks = (n + 255) / 256;  // 256 rows per block (8 waves x 32)
    sconv_wmma_kernel<<<nblocks, 256, 0, stream>>>(feat, out_pos, W, hkeys, hvals, out, n);
}